// TransformerSpatialEncoder_62947040690574
// MI455X (gfx1250) — compile-verified
//
#include <hip/hip_runtime.h>
#include <math.h>

typedef __attribute__((ext_vector_type(2))) float v2f;
typedef __attribute__((ext_vector_type(8))) float v8f;

#define HIDC 32  // per-head channel count (both layers)

// ---------------------------------------------------------------------------
// C[M,Nout] = A[M,Kin] @ W[Kin,Nout] + bias   (fp32, WMMA 16x16x4)
// one wave (32 threads) per 16x16 output tile; grid = (M/16, Nout/16)
// ---------------------------------------------------------------------------
__global__ __launch_bounds__(32)
void gemm16_wmma_f32(const float* __restrict__ A, const float* __restrict__ W,
                     const float* __restrict__ bias, float* __restrict__ C,
                     int M, int Kin, int Nout)
{
    const int lane = threadIdx.x & 31;
    const int r    = lane & 15;
    const int kh   = lane >> 4;                 // 0 or 1 (K half-select)
    const int row  = blockIdx.x * 16 + r;       // A row this lane feeds
    const int rowc = row < M ? row : M - 1;     // clamp (no divergence)
    const int col  = blockIdx.y * 16 + r;       // B/C column this lane feeds

    v8f acc = {};
    const float* ap = A + (size_t)rowc * Kin + 2 * kh;
    for (int k = 0; k < Kin; k += 4) {
        v2f a, b;
        a.x = ap[k];
        a.y = ap[k + 1];
        const float* wp = W + (size_t)(k + 2 * kh) * Nout + col;
        b.x = wp[0];
        b.y = wp[Nout];
        // (neg_a, A, neg_b, B, c_mod, C, reuse_a, reuse_b)
        acc = __builtin_amdgcn_wmma_f32_16x16x4_f32(false, a, false, b,
                                                    (short)0, acc, false, false);
    }

    const float bv    = bias[col];
    const int   rbase = blockIdx.x * 16 + kh * 8;   // VGPR j holds M=j / M=j+8
    for (int j = 0; j < 8; ++j) {
        int ro = rbase + j;
        if (ro < M) C[(size_t)ro * Nout + col] = acc[j] + bv;
    }
}

// ---------------------------------------------------------------------------
// edge-attention helpers
// ---------------------------------------------------------------------------
__device__ inline void atomicMaxF(float* addr, float v)
{
    if (v >= 0.f) atomicMax((int*)addr, __float_as_int(v));
    else          atomicMin((unsigned int*)addr, __float_as_uint(v));
}

__global__ __launch_bounds__(256)
void init_md(float* __restrict__ Mx, float* __restrict__ D, int n)
{
    int t = blockIdx.x * blockDim.x + threadIdx.x;
    if (t < n) { Mx[t] = -INFINITY; D[t] = 0.f; }
}

// pass 1: logits[e,h] = (q[dst]·k[src]) * scale ; segment max into Mx[dst,h]
__global__ __launch_bounds__(256)
void edge_logits(const float* __restrict__ q, const float* __restrict__ k,
                 const int* __restrict__ src, const int* __restrict__ dst,
                 float* __restrict__ P, float* __restrict__ Mx,
                 int E, int H, float scale)
{
    int t = blockIdx.x * blockDim.x + threadIdx.x;
    if (t >= E * H) return;
    int e = t / H, h = t - e * H;
    int s = src[e], d = dst[e];
    const float* qp = q + ((size_t)d * H + h) * HIDC;
    const float* kp = k + ((size_t)s * H + h) * HIDC;
    float acc = 0.f;
    for (int c = 0; c < HIDC; c += 4) {
        float4 qa = *(const float4*)(qp + c);
        float4 ka = *(const float4*)(kp + c);
        acc += qa.x * ka.x + qa.y * ka.y + qa.z * ka.z + qa.w * ka.w;
    }
    acc *= scale;
    P[t] = acc;
    atomicMaxF(&Mx[(size_t)d * H + h], acc);
}

// pass 2: p = exp(logit - max) ; segment sum into D[dst,h] ; P <- p
__global__ __launch_bounds__(256)
void edge_softmax(const int* __restrict__ dst, float* __restrict__ P,
                  const float* __restrict__ Mx, float* __restrict__ D,
                  int E, int H)
{
    int t = blockIdx.x * blockDim.x + threadIdx.x;
    if (t >= E * H) return;
    int e = t / H, h = t - e * H;
    int d = dst[e];
    float p = __expf(P[t] - Mx[(size_t)d * H + h]);
    P[t] = p;
    atomicAdd(&D[(size_t)d * H + h], p);
}

// pass 3: AGG[dst,h,:] += (p/denom) * V[src,h,:]   (thread = (e,h,c/4))
__global__ __launch_bounds__(256)
void edge_aggregate(const int* __restrict__ src, const int* __restrict__ dst,
                    const float* __restrict__ P, const float* __restrict__ D,
                    const float* __restrict__ V, float* __restrict__ AGG,
                    int E, int H)
{
    int t = blockIdx.x * blockDim.x + threadIdx.x;
    int total = E * H * (HIDC / 4);
    if (t >= total) return;
    int c4 = t & 7;
    int eh = t >> 3;
    int e = eh / H, h = eh - e * H;
    int s = src[e], d = dst[e];
    float alpha = P[eh] / D[(size_t)d * H + h];
    float4 v4 = *(const float4*)(V + ((size_t)s * H + h) * HIDC + c4 * 4);
    float* out = AGG + ((size_t)d * H + h) * HIDC + c4 * 4;
    atomicAdd(out + 0, alpha * v4.x);
    atomicAdd(out + 1, alpha * v4.y);
    atomicAdd(out + 2, alpha * v4.z);
    atomicAdd(out + 3, alpha * v4.w);
}

// ELU between layers
__global__ __launch_bounds__(256)
void elu_k(const float* __restrict__ in, float* __restrict__ out, int n)
{
    int t = blockIdx.x * blockDim.x + threadIdx.x;
    if (t < n) {
        float x = in[t];
        out[t] = x > 0.f ? x : expm1f(x);
    }
}

static inline unsigned cdiv(unsigned a, unsigned b) { return (a + b - 1) / b; }

extern "C" void kernel_launch(void* const* d_in, const int* in_sizes, int n_in,
                              void* d_out, int out_size, void* d_ws, size_t ws_size,
                              hipStream_t stream)
{
    const float* x   = (const float*)d_in[0];
    const int*   ei  = (const int*)d_in[1];
    const float* Wq1 = (const float*)d_in[2];  const float* bq1 = (const float*)d_in[3];
    const float* Wk1 = (const float*)d_in[4];  const float* bk1 = (const float*)d_in[5];
    const float* Wv1 = (const float*)d_in[6];  const float* bv1 = (const float*)d_in[7];
    const float* Ws1 = (const float*)d_in[8];  const float* bs1 = (const float*)d_in[9];
    const float* Wq2 = (const float*)d_in[10]; const float* bq2 = (const float*)d_in[11];
    const float* Wk2 = (const float*)d_in[12]; const float* bk2 = (const float*)d_in[13];
    const float* Wv2 = (const float*)d_in[14]; const float* bv2 = (const float*)d_in[15];
    const float* Ws2 = (const float*)d_in[16]; const float* bs2 = (const float*)d_in[17];

    const int Nn = in_sizes[0] / 64;   // 50000 nodes
    const int E  = in_sizes[1] / 2;    // 800000 edges
    const int* src = ei;
    const int* dst = ei + E;
    const float scale = 0.17677669529663689f;  // 1/sqrt(32)

    // ---- workspace layout (all fp32), ~117 MB, L2-resident working set ----
    char*  ws  = (char*)d_ws;
    size_t off = 0;
    auto  take = [&](size_t bytes) { void* p = ws + off; off = (off + bytes + 255) & ~(size_t)255; return p; };
    float* Q   = (float*)take((size_t)Nn * 128 * 4);  // layer1 q  / layer2 q
    float* K   = (float*)take((size_t)Nn * 128 * 4);  // layer1 k  / h (post-ELU)
    float* V   = (float*)take((size_t)Nn * 128 * 4);  // layer1 v  / layer2 k
    float* AGG = (float*)take((size_t)Nn * 128 * 4);  // layer1 skip+agg / layer2 v
    float* P   = (float*)take((size_t)E * 4 * 4);     // per-edge logits -> probs
    float* Mx  = (float*)take((size_t)Nn * 4 * 4);    // segment max
    float* Dd  = (float*)take((size_t)Nn * 4 * 4);    // segment denom
    float* Hb  = K;                                   // ELU output aliases K
    float* OUT = (float*)d_out;
    (void)ws_size; (void)n_in; (void)out_size;

    const unsigned mt = cdiv(Nn, 16);

    // ================= layer 1: IN=64 -> HEADS*HID=128 =================
    {
        dim3 g(mt, 128 / 16);
        gemm16_wmma_f32<<<g, 32, 0, stream>>>(x, Wq1, bq1, Q,   Nn, 64, 128);
        gemm16_wmma_f32<<<g, 32, 0, stream>>>(x, Wk1, bk1, K,   Nn, 64, 128);
        gemm16_wmma_f32<<<g, 32, 0, stream>>>(x, Wv1, bv1, V,   Nn, 64, 128);
        gemm16_wmma_f32<<<g, 32, 0, stream>>>(x, Ws1, bs1, AGG, Nn, 64, 128); // skip pre-loaded
    }
    init_md<<<cdiv(Nn * 4, 256), 256, 0, stream>>>(Mx, Dd, Nn * 4);
    edge_logits   <<<cdiv(E * 4, 256), 256, 0, stream>>>(Q, K, src, dst, P, Mx, E, 4, scale);
    edge_softmax  <<<cdiv(E * 4, 256), 256, 0, stream>>>(dst, P, Mx, Dd, E, 4);
    edge_aggregate<<<cdiv(E * 4 * 8, 256), 256, 0, stream>>>(src, dst, P, Dd, V, AGG, E, 4);
    elu_k<<<cdiv(Nn * 128, 256), 256, 0, stream>>>(AGG, Hb, Nn * 128);

    // ================= layer 2: 128 -> 32, heads=1 =====================
    {
        dim3 g(mt, 32 / 16);
        gemm16_wmma_f32<<<g, 32, 0, stream>>>(Hb, Wq2, bq2, Q,   Nn, 128, 32);
        gemm16_wmma_f32<<<g, 32, 0, stream>>>(Hb, Wk2, bk2, V,   Nn, 128, 32);
        gemm16_wmma_f32<<<g, 32, 0, stream>>>(Hb, Wv2, bv2, AGG, Nn, 128, 32);
        gemm16_wmma_f32<<<g, 32, 0, stream>>>(Hb, Ws2, bs2, OUT, Nn, 128, 32); // skip -> d_out
    }
    init_md<<<cdiv(Nn, 256), 256, 0, stream>>>(Mx, Dd, Nn);
    edge_logits   <<<cdiv(E, 256), 256, 0, stream>>>(Q, V, src, dst, P, Mx, E, 1, scale);
    edge_softmax  <<<cdiv(E, 256), 256, 0, stream>>>(dst, P, Mx, Dd, E, 1);
    edge_aggregate<<<cdiv(E * 8, 256), 256, 0, stream>>>(src, dst, P, Dd, AGG, OUT, E, 1);
    // heads=1, concat=False -> mean == the single head; skip already in OUT. Done.
}